// MaxPoolLocal_30021821399291
// MI455X (gfx1250) — compile-verified
//
#include <hip/hip_runtime.h>
#include <stdint.h>

// Problem constants (match reference): B=8, F=128, N_IN=50000, N_OUT=25000, K=16
#define C_   1024        // B*F channels
#define NIN  50000
#define NOUT 25000
#define K_   16

// ---------------------------------------------------------------------------
// CDNA5 async global->LDS helpers (ASYNCcnt path).
// Builtin signature (from hipcc diagnostic): first arg is addrspace(1)
// (printed "__device__") non-const int4-vector pointer; LDS side addrspace(3).
// ---------------------------------------------------------------------------
typedef int v4i __attribute__((vector_size(16)));
typedef __attribute__((address_space(1))) v4i* g4p_t;
typedef __attribute__((address_space(3))) v4i* l4p_t;

static __device__ __forceinline__ void async_ld16(const float* g, float* l) {
#if __has_builtin(__builtin_amdgcn_global_load_async_to_lds_b128)
  __builtin_amdgcn_global_load_async_to_lds_b128((g4p_t)g, (l4p_t)l, 0, 0);
#else
  // lds byte offset (addrspace(3) pointers are 32-bit LDS offsets)
  unsigned lofs = (unsigned)(size_t)(__attribute__((address_space(3))) float*)l;
  asm volatile("global_load_async_to_lds_b128 %0, %1, off"
               :: "v"(lofs), "v"(g) : "memory");
#endif
}

static __device__ __forceinline__ void wait_async0() {
#if __has_builtin(__builtin_amdgcn_s_wait_asynccnt)
  __builtin_amdgcn_s_wait_asynccnt(0);
#else
  asm volatile("s_wait_asynccnt 0x0" ::: "memory");
#endif
}

// ---------------------------------------------------------------------------
// Pass 1: transpose x[C][NIN] -> xT[NIN][C] using async global->LDS staging.
// 64(i) x 64(c) tiles; loads are b128 async straight into LDS (no VGPR
// round-trip), drained with s_wait_asynccnt; stores are 256B-contiguous
// float4 runs along c. Row stride 68 floats keeps b128 LDS writes 16B-aligned.
// NIN % 4 == 0, so every b128 in the ragged last i-tile is still full.
// ---------------------------------------------------------------------------
#define TI 64
#define TC 64
#define LSTR 68   // floats per c-row in LDS (64 + 4 pad, 272B = 16B aligned)

__global__ __launch_bounds__(256) void xpose_kernel(const float* __restrict__ x,
                                                    float* __restrict__ xT) {
  __shared__ float tile[TC * LSTR];       // 17408 B
  const int tid = threadIdx.x;            // 0..255 (8 waves)
  const int i0 = blockIdx.x * TI;
  const int c0 = blockIdx.y * TC;

  // Load phase: 64 c-rows x 16 float4 along i -> 1024 async b128s per block.
#pragma unroll
  for (int j = 0; j < 4; ++j) {
    const int q  = tid + 256 * j;         // 0..1023
    const int cl = q >> 4;                // 0..63
    const int i4 = q & 15;                // 0..15
    const int i  = i0 + i4 * 4;
    if (i < NIN)
      async_ld16(x + (size_t)(c0 + cl) * NIN + i, &tile[cl * LSTR + i4 * 4]);
  }
  wait_async0();       // this wave's ASYNCcnt == 0
  __syncthreads();     // cross-wave visibility of LDS tile

  // Store phase: 64 i-rows x 16 float4 along c (coalesced 256B per i-row).
#pragma unroll
  for (int j = 0; j < 4; ++j) {
    const int q  = tid + 256 * j;
    const int il = q >> 4;                // 0..63
    const int cb = (q & 15) * 4;          // 0,4,..,60
    const int i  = i0 + il;
    if (i < NIN) {
      float4 v;
      v.x = tile[(cb + 0) * LSTR + il];
      v.y = tile[(cb + 1) * LSTR + il];
      v.z = tile[(cb + 2) * LSTR + il];
      v.w = tile[(cb + 3) * LSTR + il];
      *(float4*)(xT + (size_t)i * C_ + c0 + cb) = v;
    }
  }
}

// ---------------------------------------------------------------------------
// Pass 2: gather + max. One block = 16 output points x all 1024 channels.
// Thread t owns channels 4t..4t+3 (float4): a wave's gather of row xT[idx]
// touches consecutive 128B lines with every byte used (zero divergence).
// Indices are block-uniform -> scalar loads into SGPRs.
// Results staged in 64KB LDS (sh[n][c]) then written as coalesced float4
// runs along n (out layout is (c, n)).
// ---------------------------------------------------------------------------
__global__ __launch_bounds__(256) void gathermax_kernel(const float* __restrict__ xT,
                                                        const uint32_t* __restrict__ nbr2, // int64 viewed as u32 pairs
                                                        float* __restrict__ out) {
  __shared__ float sh[16 * C_];           // 64 KB: sh[nLocal*1024 + c]
  const int t  = threadIdx.x;             // 0..255
  const int n0 = blockIdx.x * 16;
  const float4* __restrict__ xT4 = (const float4*)xT;
  float4* __restrict__ sh4 = (float4*)sh;

  const int nrem = (NOUT - n0) < 16 ? (NOUT - n0) : 16;

  for (int ni = 0; ni < nrem; ++ni) {
    const int n = n0 + ni;

    // Prefetch next point's index line (uniform -> global_prefetch_b8)
    if (ni + 1 < nrem)
      __builtin_prefetch(&nbr2[((size_t)(n + 1) * K_) * 2], 0, 0);

    // 16 block-uniform indices: low dword of each little-endian int64
    uint32_t idx[K_];
#pragma unroll
    for (int k = 0; k < K_; ++k)
      idx[k] = nbr2[((size_t)n * K_ + k) * 2];

    // Issue all 16 independent b128 gathers first (MLP), then tree-max.
    float4 v[K_];
#pragma unroll
    for (int k = 0; k < K_; ++k)
      v[k] = xT4[(size_t)idx[k] * (C_ / 4) + t];

#pragma unroll
    for (int s = K_ / 2; s >= 1; s >>= 1) {
#pragma unroll
      for (int k = 0; k < s; ++k) {
        v[k].x = fmaxf(v[k].x, v[k + s].x);
        v[k].y = fmaxf(v[k].y, v[k + s].y);
        v[k].z = fmaxf(v[k].z, v[k + s].z);
        v[k].w = fmaxf(v[k].w, v[k + s].w);
      }
    }
    sh4[ni * (C_ / 4) + t] = v[0];
  }
  __syncthreads();

  // Write-out: out[c*NOUT + n0 .. n0+15]. (c*NOUT + n0)*4 is 16B-aligned
  // (25000 % 4 == 0, n0 % 16 == 0) -> float4 stores, full/half L2 lines.
  if (nrem == 16) {
    for (int c = t; c < C_; c += 256) {
      float vv[16];
#pragma unroll
      for (int nI = 0; nI < 16; ++nI) vv[nI] = sh[nI * C_ + c];
      float4* dst = (float4*)(out + (size_t)c * NOUT + n0);
      dst[0] = make_float4(vv[0],  vv[1],  vv[2],  vv[3]);
      dst[1] = make_float4(vv[4],  vv[5],  vv[6],  vv[7]);
      dst[2] = make_float4(vv[8],  vv[9],  vv[10], vv[11]);
      dst[3] = make_float4(vv[12], vv[13], vv[14], vv[15]);
    }
  } else {
    for (int c = t; c < C_; c += 256)
      for (int nI = 0; nI < nrem; ++nI)
        out[(size_t)c * NOUT + n0 + nI] = sh[nI * C_ + c];
  }
}

// ---------------------------------------------------------------------------
// Fallback (only if workspace can't hold the 204.8 MB transposed tensor):
// direct divergent gather — correct but slow.
// ---------------------------------------------------------------------------
__global__ __launch_bounds__(256) void direct_kernel(const float* __restrict__ x,
                                                     const uint32_t* __restrict__ nbr2,
                                                     float* __restrict__ out) {
  const int n = blockIdx.x;
  uint32_t idx[K_];
#pragma unroll
  for (int k = 0; k < K_; ++k)
    idx[k] = nbr2[((size_t)n * K_ + k) * 2];
  for (int c = threadIdx.x; c < C_; c += 256) {
    const float* row = x + (size_t)c * NIN;
    float m = row[idx[0]];
#pragma unroll
    for (int k = 1; k < K_; ++k) m = fmaxf(m, row[idx[k]]);
    out[(size_t)c * NOUT + n] = m;
  }
}

extern "C" void kernel_launch(void* const* d_in, const int* in_sizes, int n_in,
                              void* d_out, int out_size, void* d_ws, size_t ws_size,
                              hipStream_t stream) {
  const float*    x    = (const float*)d_in[0];
  const uint32_t* nbr2 = (const uint32_t*)d_in[1];   // int64 indices, read low dwords
  float*          out  = (float*)d_out;

  const size_t need = (size_t)NIN * C_ * sizeof(float);  // 204.8 MB transposed copy
  if (ws_size >= need) {
    float* xT = (float*)d_ws;
    dim3 tb(256, 1);
    dim3 tg((NIN + TI - 1) / TI, C_ / TC);               // 782 x 16
    xpose_kernel<<<tg, tb, 0, stream>>>(x, xT);

    const int nblocks = (NOUT + 15) / 16;                // 1563 blocks x 8 waves
    gathermax_kernel<<<nblocks, 256, 0, stream>>>(xT, nbr2, out);
  } else {
    direct_kernel<<<NOUT, 256, 0, stream>>>(x, nbr2, out);
  }
}